// MCCALossID_85813446574618
// MI455X (gfx1250) — compile-verified
//
#include <hip/hip_runtime.h>
#include <hip/hip_bf16.h>

typedef __attribute__((ext_vector_type(2))) float v2f;
typedef __attribute__((ext_vector_type(8))) float v8f;

#define NUM_CLASSES 395
#define CPAD        400      // 25 tiles of 16
#define FEAT        2048
#define BATCH       16384
#define RHO_C       4500.0f
#define SIGMA_C     8.0f
#define CLIP_LO     1e-12f
#define CLIP_HI     1e12f
#define INF_BITS    0x7F800000u

// ---------------------------------------------------------------------------
// Init: zero-row for padded tile lanes, accumulators, per-class min slots.
// ---------------------------------------------------------------------------
__global__ void mcca_init_kernel(float* __restrict__ zrow,
                                 unsigned* __restrict__ perclass,
                                 float* __restrict__ accum) {
    int i = threadIdx.x;                    // 1024 threads
    for (int j = i; j < FEAT; j += 1024) zrow[j] = 0.0f;
    if (i < CPAD) perclass[i] = INF_BITS;
    if (i == 0) { accum[0] = 0.0f; accum[1] = 0.0f; }
}

// ---------------------------------------------------------------------------
// Gram matrix G = centers @ centers^T via V_WMMA_F32_16X16X4_F32.
// One wave32 per 16x16 tile; K-loop over 2048 in steps of 4.
// Out-of-range rows read a zero row => no EXEC predication in the hot loop,
// loads are unconditional global_load_b64 with immediate offsets.
// Two accumulators hide the WMMA D->C dependency latency.
// ---------------------------------------------------------------------------
__global__ void mcca_gram_wmma_kernel(const float* __restrict__ centers,
                                      const float* __restrict__ zrow,
                                      float* __restrict__ G) {
    const int ti   = blockIdx.x;
    const int tj   = blockIdx.y;
    const int lane = threadIdx.x & 31;
    const int half = lane >> 4;     // 0: K offset 0, 1: K offset 2
    const int l16  = lane & 15;

    const int row_m = ti * 16 + l16;   // A-side center row
    const int col_n = tj * 16 + l16;   // B-side center row

    const float* ap = (row_m < NUM_CLASSES)
                        ? centers + (size_t)row_m * FEAT + 2 * half
                        : zrow + 2 * half;
    const float* bp = (col_n < NUM_CLASSES)
                        ? centers + (size_t)col_n * FEAT + 2 * half
                        : zrow + 2 * half;

    v8f c0 = {};
    v8f c1 = {};
#pragma unroll 4
    for (int kk = 0; kk < FEAT; kk += 8) {
        v2f a0 = *(const v2f*)(ap + kk);
        v2f b0 = *(const v2f*)(bp + kk);
        v2f a1 = *(const v2f*)(ap + kk + 4);
        v2f b1 = *(const v2f*)(bp + kk + 4);
        c0 = __builtin_amdgcn_wmma_f32_16x16x4_f32(
                false, a0, false, b0, (short)0, c0, false, false);
        c1 = __builtin_amdgcn_wmma_f32_16x16x4_f32(
                false, a1, false, b1, (short)0, c1, false, false);
    }

    const int gn = tj * 16 + l16;
#pragma unroll
    for (int v = 0; v < 8; ++v) {
        int gm = ti * 16 + v + 8 * half;   // C/D layout: lanes 16-31 -> M+8
        G[(size_t)gm * CPAD + gn] = c0[v] + c1[v];
    }
}

// ---------------------------------------------------------------------------
// minval[k] = min_{c != k} ( G[k,k] + G[c,c] - 2*G[k,c] )
// (c_sq is the diagonal of G). 395 threads, trivial; G rows are L2-hot.
// ---------------------------------------------------------------------------
__global__ void mcca_minval_kernel(const float* __restrict__ G,
                                   float* __restrict__ minval) {
    int k = blockIdx.x * blockDim.x + threadIdx.x;
    if (k >= NUM_CLASSES) return;
    const float csk = G[(size_t)k * CPAD + k];
    float m = 3.402823466e38f;
    for (int c = 0; c < NUM_CLASSES; ++c) {
        if (c == k) continue;
        float v = csk + G[(size_t)c * CPAD + c] - 2.0f * G[(size_t)k * CPAD + c];
        m = fminf(m, v);
    }
    minval[k] = m;
}

// ---------------------------------------------------------------------------
// Streaming pass over x (the only HBM-bound work: 134 MB ~ 5.8 us @ 23.3 TB/s).
// One wave32 per row: 16 float4 loads/lane of x and of the gathered center
// (centers are L2-resident), d = ||x - c||^2, cross-lane reduce, then
// block-combined atomics: sum of clip(d), sum of clip(minval[lab]),
// per-class atomic-min of d (d >= 0 so uint bit order == float order).
// ---------------------------------------------------------------------------
__global__ void mcca_dist_kernel(const float* __restrict__ x,
                                 const float* __restrict__ centers,
                                 const int* __restrict__ labels,
                                 const float* __restrict__ minval,
                                 unsigned* __restrict__ perclass,
                                 float* __restrict__ accum) {
    const int wave = threadIdx.x >> 5;
    const int lane = threadIdx.x & 31;
    const int row  = blockIdx.x * 8 + wave;

    const int lab = labels[row];
    const float4* xr = (const float4*)(x + (size_t)row * FEAT);
    const float4* cr = (const float4*)(centers + (size_t)lab * FEAT);

    float acc = 0.0f;
#pragma unroll
    for (int k = 0; k < 16; ++k) {
        float4 xv = xr[k * 32 + lane];
        float4 cv = cr[k * 32 + lane];
        float d0 = xv.x - cv.x;
        float d1 = xv.y - cv.y;
        float d2 = xv.z - cv.z;
        float d3 = xv.w - cv.w;
        acc = fmaf(d0, d0, acc);
        acc = fmaf(d1, d1, acc);
        acc = fmaf(d2, d2, acc);
        acc = fmaf(d3, d3, acc);
    }
#pragma unroll
    for (int off = 16; off > 0; off >>= 1)
        acc += __shfl_xor(acc, off, 32);

    __shared__ float s_d[8];
    __shared__ float s_v[8];
    if (lane == 0) {
        const float d = acc;
        s_d[wave] = fminf(fmaxf(d, CLIP_LO), CLIP_HI);
        s_v[wave] = fminf(fmaxf(minval[lab], CLIP_LO), CLIP_HI);
        if (d != 0.0f)
            atomicMin(&perclass[lab], __float_as_uint(d));
    }
    __syncthreads();
    if (threadIdx.x == 0) {
        float sd = 0.0f, sv = 0.0f;
#pragma unroll
        for (int i = 0; i < 8; ++i) { sd += s_d[i]; sv += s_v[i]; }
        atomicAdd(&accum[0], sd);
        atomicAdd(&accum[1], sv);
    }
}

// ---------------------------------------------------------------------------
// Final scalar combine (395-element loop, single thread).
// sum(clip(dist)) over the full B x C matrix = sum(clip(d)) + B*(C-1)*1e-12.
// ---------------------------------------------------------------------------
__global__ void mcca_final_kernel(const unsigned* __restrict__ perclass,
                                  const float* __restrict__ accum,
                                  float* __restrict__ out) {
    float nearest = 0.0f;
    for (int c = 0; c < NUM_CLASSES; ++c) {
        unsigned b = perclass[c];
        if (b != INF_BITS) nearest += __uint_as_float(b);
    }
    const float invB = 1.0f / (float)BATCH;
    float sum_d = accum[0] + (float)BATCH * (float)(NUM_CLASSES - 1) * CLIP_LO;
    float loss_center  = fmaxf(sum_d * invB, 0.0f);
    float loss_nearest = nearest * 8.0f * invB;
    float loss_push    = fmaxf(RHO_C - accum[1] * invB, 0.0f);
    float loss_1       = fmaxf(loss_center - loss_nearest - SIGMA_C, 0.0f);
    out[0] = loss_1 + loss_push;
}

// ---------------------------------------------------------------------------
extern "C" void kernel_launch(void* const* d_in, const int* in_sizes, int n_in,
                              void* d_out, int out_size, void* d_ws, size_t ws_size,
                              hipStream_t stream) {
    const float* x       = (const float*)d_in[0];  // [BATCH, FEAT]
    const float* centers = (const float*)d_in[1];  // [NUM_CLASSES, FEAT]
    const int*   labels  = (const int*)d_in[2];    // [BATCH]
    float* out = (float*)d_out;

    // Workspace layout
    float*    zrow     = (float*)d_ws;                 // FEAT floats (zero row)
    float*    G        = zrow + FEAT;                  // CPAD*CPAD floats (640 KB)
    float*    minval   = G + (size_t)CPAD * CPAD;      // CPAD floats
    unsigned* perclass = (unsigned*)(minval + CPAD);   // CPAD uints
    float*    accum    = (float*)(perclass + CPAD);    // [0]=sum clip(d), [1]=sum clip(val)

    mcca_init_kernel<<<1, 1024, 0, stream>>>(zrow, perclass, accum);

    dim3 gtiles(CPAD / 16, CPAD / 16);                 // 25 x 25 tiles
    mcca_gram_wmma_kernel<<<gtiles, 32, 0, stream>>>(centers, zrow, G);

    mcca_minval_kernel<<<(NUM_CLASSES + 127) / 128, 128, 0, stream>>>(G, minval);

    mcca_dist_kernel<<<BATCH / 8, 256, 0, stream>>>(x, centers, labels, minval,
                                                    perclass, accum);

    mcca_final_kernel<<<1, 1, 0, stream>>>(perclass, accum, out);
}